// SetAbstraction_11407433138469
// MI455X (gfx1250) — compile-verified
//
#include <hip/hip_runtime.h>
#include <hip/hip_bf16.h>

typedef __attribute__((ext_vector_type(2))) float v2f;
typedef __attribute__((ext_vector_type(8))) float v8f;

namespace {
constexpr int B_    = 4;
constexpr int N_    = 65536;
constexpr int M_    = 16384;
constexpr int K_    = 32;
constexpr int CIN   = 64;
constexpr int COUT  = 128;
constexpr int HID   = 64;
constexpr int PE_   = 16;
constexpr int GIN   = 83;
constexpr int GINP  = 84;          // padded K-dim for matmul0
constexpr int PTS   = 4;           // points per workgroup
constexpr int ROWS  = PTS * K_;    // 128 rows of grouped matrix
constexpr int BLOCK = 256;         // 8 waves (wave32)

// Padded LDS row strides (floats) chosen for 64-bank conflict freedom:
//  - SGS=85  : 21*r mod 64 distinct for r=0..15; +2 half-wave shift collides for dr=6 (2-way)
//  - SHS=69  : 5*r  mod 64 distinct; +2 shift needs dr=26 -> never collides
//  - SOS=130 : 2*t  mod 64 distinct for 32 lanes; 8-row C-store halves 16 banks apart
//  - W0S=72, W1S=136 : half-wave B-fragment rows land 16 banks apart (disjoint)
constexpr int SGS = 85;
constexpr int SHS = 69;
constexpr int SOS = 130;
constexpr int W0S = 72;
constexpr int W1S = 136;

// LDS layout in floats (dynamic LDS, ~158 KB -> 2 WGs per 320 KB WGP)
constexpr int OFF_A  = 0;                         // sG [128][85] then sOut [128][130]
constexpr int SZ_A   = ROWS * SOS;                // 16640 (covers 128*85=10880)
constexpr int OFF_H  = OFF_A + SZ_A;              // sH [128][69] then sRes [4][128]
constexpr int SZ_H   = ROWS * SHS;                // 8832
constexpr int OFF_W0 = OFF_H + SZ_H;              // W0 padded [84][72]
constexpr int SZ_W0  = GINP * W0S;                // 6048
constexpr int OFF_W1 = OFF_W0 + SZ_W0;            // W1 [64][136], later rw [64][136]
constexpr int SZ_W1  = HID * W1S;                 // 8704
constexpr int OFF_ID = OFF_W1 + SZ_W1;            // identity feats [4][64]
constexpr int SZ_ID  = PTS * CIN;                 // 256
constexpr int OFF_RS = OFF_ID + SZ_ID;            // residual mean/rstd [2][4]
constexpr int LDS_FLOATS = OFF_RS + 2 * PTS;      // 40488 floats = 161952 B
}

__device__ __forceinline__ float gelu_exact(float x) {
    return 0.5f * x * (1.0f + erff(x * 0.70710678118654752440f));
}

__global__ void __launch_bounds__(BLOCK)
set_abstraction_kernel(const float* __restrict__ position,
                       const float* __restrict__ feature,
                       const int*   __restrict__ sample_idx,
                       const int*   __restrict__ neighbor_idx,
                       const float* __restrict__ w0,  const float* __restrict__ b0,
                       const float* __restrict__ g0,  const float* __restrict__ bt0,
                       const float* __restrict__ w1,  const float* __restrict__ b1,
                       const float* __restrict__ g1,  const float* __restrict__ bt1,
                       const float* __restrict__ pe_w, const float* __restrict__ pe_b,
                       const float* __restrict__ rw,  const float* __restrict__ rb,
                       const float* __restrict__ rg,  const float* __restrict__ rbt,
                       float* __restrict__ out_pos,   // [B,M,3]
                       float* __restrict__ out_feat)  // [B,M,128]
{
    extern __shared__ float sm[];
    const int tid = threadIdx.x;
    const int gp  = blockIdx.x * PTS;   // first global point index of this WG

    // ---------------- Phase A: load weights, gather grouped matrix ----------------
    for (int i = tid; i < GINP * HID; i += BLOCK) {
        int r = i >> 6, c = i & 63;
        sm[OFF_W0 + r * W0S + c] = (r < GIN) ? w0[i] : 0.0f;   // zero-pad row 83
    }
    for (int i = tid; i < HID * COUT; i += BLOCK) {
        int r = i >> 7, c = i & 127;
        sm[OFF_W1 + r * W1S + c] = w1[i];
    }
    for (int i = tid; i < SZ_ID; i += BLOCK) {
        int p = i >> 6, j = i & 63;
        int gpi = gp + p, b = gpi / M_, m = gpi % M_;
        int sidx = sample_idx[b * M_ + m];
        sm[OFF_ID + i] = feature[(b * N_ + sidx) * CIN + j];
    }
    {   // 2 threads per grouped row: half the features + half the PE each
        int r = tid >> 1, h = tid & 1;
        int p = r >> 5, k = r & 31;
        int gpi = gp + p, b = gpi / M_, m = gpi % M_;
        int sidx = sample_idx[b * M_ + m];
        int nidx = neighbor_idx[(b * M_ + m) * K_ + k];
        const float* sp = &position[(b * N_ + sidx) * 3];
        const float* np = &position[(b * N_ + nidx) * 3];
        float rx = np[0] - sp[0], ry = np[1] - sp[1], rz = np[2] - sp[2];
        float* row = &sm[OFF_A + r * SGS];
        const float4* f4 = (const float4*)&feature[(b * N_ + nidx) * CIN];
        if (h == 0) {
            row[0] = rx; row[1] = ry; row[2] = rz;
#pragma unroll
            for (int i = 0; i < 8; ++i) {
                float4 v = f4[i];
                row[3 + 4 * i + 0] = v.x; row[3 + 4 * i + 1] = v.y;
                row[3 + 4 * i + 2] = v.z; row[3 + 4 * i + 3] = v.w;
            }
#pragma unroll
            for (int j = 0; j < 8; ++j)
                row[3 + CIN + j] = pe_b[j] + rx * pe_w[j] + ry * pe_w[PE_ + j] + rz * pe_w[2 * PE_ + j];
        } else {
#pragma unroll
            for (int i = 8; i < 16; ++i) {
                float4 v = f4[i];
                row[3 + 4 * i + 0] = v.x; row[3 + 4 * i + 1] = v.y;
                row[3 + 4 * i + 2] = v.z; row[3 + 4 * i + 3] = v.w;
            }
#pragma unroll
            for (int j = 8; j < 16; ++j)
                row[3 + CIN + j] = pe_b[j] + rx * pe_w[j] + ry * pe_w[PE_ + j] + rz * pe_w[2 * PE_ + j];
            row[GIN] = 0.0f;                                   // pad col 83
        }
    }
    __syncthreads();

    const int wave  = tid >> 5;
    const int lane  = tid & 31;
    const int lrow  = lane & 15;           // row/col within tile for this lane
    const int khalf = (lane >> 4) << 1;    // lanes 16-31 carry K+2,K+3
    const int rhalf = (lane >> 4) << 3;    // C/D: lanes 16-31 carry rows M+8..M+15

    // ---------------- Phase B: matmul0  [128x84] @ [84x64] -> sH ----------------
    {
        v8f acc[4];
#pragma unroll
        for (int j = 0; j < 4; ++j) acc[j] = v8f{0.f,0.f,0.f,0.f,0.f,0.f,0.f,0.f};
        const float* sG  = &sm[OFF_A];
        const float* sW0 = &sm[OFF_W0];
        const int arow = 16 * wave + lrow;
        for (int ks = 0; ks < GINP; ks += 4) {
            v2f a;
            const float* ap = sG + arow * SGS + ks + khalf;
            a.x = ap[0]; a.y = ap[1];
#pragma unroll
            for (int j = 0; j < 4; ++j) {
                v2f bf;
                const float* bp = sW0 + (ks + khalf) * W0S + 16 * j + lrow;
                bf.x = bp[0]; bf.y = bp[W0S];
                acc[j] = __builtin_amdgcn_wmma_f32_16x16x4_f32(
                    false, a, false, bf, (short)0, acc[j], false, false);
            }
        }
        float* sH = &sm[OFF_H];
#pragma unroll
        for (int j = 0; j < 4; ++j) {
            int col = 16 * j + lrow;
            float bv = b0[col];
#pragma unroll
            for (int v = 0; v < 8; ++v) {
                int row = 16 * wave + rhalf + v;
                sH[row * SHS + col] = acc[j][v] + bv;
            }
        }
    }
    __syncthreads();

    // ---------------- Phase B2: LayerNorm0 + exact GELU (per row of 64) ----------------
    if (tid < ROWS) {
        float* row = &sm[OFF_H + tid * SHS];
        float mu = 0.f;
        for (int c = 0; c < HID; ++c) mu += row[c];
        mu *= (1.0f / HID);
        float var = 0.f;
        for (int c = 0; c < HID; ++c) { float d = row[c] - mu; var += d * d; }
        var *= (1.0f / HID);
        float rinv = rsqrtf(var + 1e-5f);
        for (int c = 0; c < HID; ++c)
            row[c] = gelu_exact((row[c] - mu) * rinv * g0[c] + bt0[c]);
    }
    __syncthreads();

    // ---------------- Phase C: matmul1  [128x64] @ [64x128] -> sOut (reuses sG space) ----
    {
        v8f acc[8];
#pragma unroll
        for (int j = 0; j < 8; ++j) acc[j] = v8f{0.f,0.f,0.f,0.f,0.f,0.f,0.f,0.f};
        const float* sH  = &sm[OFF_H];
        const float* sW1 = &sm[OFF_W1];
        const int arow = 16 * wave + lrow;
        for (int ks = 0; ks < HID; ks += 4) {
            v2f a;
            const float* ap = sH + arow * SHS + ks + khalf;
            a.x = ap[0]; a.y = ap[1];
#pragma unroll
            for (int j = 0; j < 8; ++j) {
                v2f bf;
                const float* bp = sW1 + (ks + khalf) * W1S + 16 * j + lrow;
                bf.x = bp[0]; bf.y = bp[W1S];
                acc[j] = __builtin_amdgcn_wmma_f32_16x16x4_f32(
                    false, a, false, bf, (short)0, acc[j], false, false);
            }
        }
        float* sOut = &sm[OFF_A];
#pragma unroll
        for (int j = 0; j < 8; ++j) {
            int col = 16 * j + lrow;
            float bv = b1[col];
#pragma unroll
            for (int v = 0; v < 8; ++v) {
                int row = 16 * wave + rhalf + v;
                sOut[row * SOS + col] = acc[j][v] + bv;
            }
        }
    }
    __syncthreads();

    // ------- Phase D1: LayerNorm1 (per row of 128) + stage rw into dead W1 region -------
    if (tid < ROWS) {
        float* row = &sm[OFF_A + tid * SOS];
        float mu = 0.f;
        for (int c = 0; c < COUT; ++c) mu += row[c];
        mu *= (1.0f / COUT);
        float var = 0.f;
        for (int c = 0; c < COUT; ++c) { float d = row[c] - mu; var += d * d; }
        var *= (1.0f / COUT);
        float rinv = rsqrtf(var + 1e-5f);
        for (int c = 0; c < COUT; ++c)
            row[c] = (row[c] - mu) * rinv * g1[c] + bt1[c];
    }
    for (int i = tid; i < HID * COUT; i += BLOCK) {   // coalesced rw -> LDS
        int r = i >> 7, c = i & 127;
        sm[OFF_W1 + r * W1S + c] = rw[i];
    }
    __syncthreads();

    // ---------------- Phase D2: residual projection identity @ rw + rb -> sRes ----------
    for (int idx = tid; idx < PTS * COUT; idx += BLOCK) {
        int p = idx >> 7, c = idx & 127;
        const float* id = &sm[OFF_ID + p * CIN];
        float accv = rb[c];
        for (int j = 0; j < CIN; ++j) accv += id[j] * sm[OFF_W1 + j * W1S + c];
        sm[OFF_H + p * COUT + c] = accv;          // sRes lives in sH space
    }
    __syncthreads();

    // ---------------- Phase D3: residual LayerNorm stats ----------------
    if (tid < PTS) {
        const float* res = &sm[OFF_H + tid * COUT];
        float mu = 0.f;
        for (int c = 0; c < COUT; ++c) mu += res[c];
        mu *= (1.0f / COUT);
        float var = 0.f;
        for (int c = 0; c < COUT; ++c) { float d = res[c] - mu; var += d * d; }
        var *= (1.0f / COUT);
        sm[OFF_RS + tid]       = mu;
        sm[OFF_RS + PTS + tid] = rsqrtf(var + 1e-5f);
    }
    __syncthreads();

    // ---------------- Phase D4: max-pool over K + residual + GELU, write outputs --------
    for (int idx = tid; idx < PTS * COUT; idx += BLOCK) {
        int p = idx >> 7, c = idx & 127;
        int gpi = gp + p;
        const float* sOut = &sm[OFF_A];
        float pooled = -3.402823466e+38f;
        for (int k = 0; k < K_; ++k)
            pooled = fmaxf(pooled, sOut[(p * K_ + k) * SOS + c]);
        float mu   = sm[OFF_RS + p];
        float rinv = sm[OFF_RS + PTS + p];
        float resn = (sm[OFF_H + p * COUT + c] - mu) * rinv * rg[c] + rbt[c];
        out_feat[gpi * COUT + c] = gelu_exact(pooled + resn);
    }
    if (tid < PTS * 3) {
        int p = tid / 3, d = tid % 3;
        int gpi = gp + p, b = gpi / M_, m = gpi % M_;
        int sidx = sample_idx[b * M_ + m];
        out_pos[gpi * 3 + d] = position[(b * N_ + sidx) * 3 + d];
    }
}

extern "C" void kernel_launch(void* const* d_in, const int* in_sizes, int n_in,
                              void* d_out, int out_size, void* d_ws, size_t ws_size,
                              hipStream_t stream) {
    const float* position     = (const float*)d_in[0];
    const float* feature      = (const float*)d_in[1];
    const int*   sample_idx   = (const int*)d_in[2];
    const int*   neighbor_idx = (const int*)d_in[3];
    const float* w0  = (const float*)d_in[4];
    const float* b0  = (const float*)d_in[5];
    const float* g0  = (const float*)d_in[6];
    const float* bt0 = (const float*)d_in[7];
    const float* w1  = (const float*)d_in[8];
    const float* b1  = (const float*)d_in[9];
    const float* g1  = (const float*)d_in[10];
    const float* bt1 = (const float*)d_in[11];
    const float* pe_w = (const float*)d_in[12];
    const float* pe_b = (const float*)d_in[13];
    const float* rw  = (const float*)d_in[14];
    const float* rb  = (const float*)d_in[15];
    const float* rg  = (const float*)d_in[16];
    const float* rbt = (const float*)d_in[17];

    float* out_pos  = (float*)d_out;                 // [B,M,3] first
    float* out_feat = out_pos + (size_t)B_ * M_ * 3; // then [B,M,128]

    dim3 grid(B_ * M_ / PTS);   // 16384 workgroups
    dim3 block(BLOCK);
    size_t lds_bytes = (size_t)LDS_FLOATS * sizeof(float);
    set_abstraction_kernel<<<grid, block, lds_bytes, stream>>>(
        position, feature, sample_idx, neighbor_idx,
        w0, b0, g0, bt0, w1, b1, g1, bt1,
        pe_w, pe_b, rw, rb, rg, rbt,
        out_pos, out_feat);
}